// BistableHypergraphSynapse_65369402245523
// MI455X (gfx1250) — compile-verified
//
#include <hip/hip_runtime.h>

// ---------------------------------------------------------------------------
// BistableHypergraphSynapse for MI455X (gfx1250, wave32)
//
// Factorization: data[c,b,o] = x[b,node[c]] * cw[c] * W[o]  ->  channel dim
// pulled out of both scatter passes; epilogue is a rank-1 outer product done
// with V_WMMA_F32_16X16X4_F32 (exact f32, bias in the C accumulator).
// Scratch (~5.6 MB) and the 12.8 MB index array are L2-resident (192 MB L2);
// limiter is L2 f32-atomic throughput on the two scatter passes.
// ---------------------------------------------------------------------------

#define NUM_NEURONS 100000
#define MAX_EDGES   100000
#define NUM_CONN    1600000
#define BATCH       4
#define OUT_CH      16

typedef __attribute__((ext_vector_type(2))) float v2f;
typedef __attribute__((ext_vector_type(8))) float v8f;

// workspace layout (float offsets); all regions 16-byte aligned
#define WS_EFFW 0u        // [0      , 100000)  effective_w = s_e * w_hat
#define WS_S    100000u   // [100000 , 500000)  S then T (E x 4)
#define WS_CNT  500000u   // [500000 , 600000)  per-edge counts (float)
#define WS_U    600000u   // [600000 ,1000000)  U (N x 4)
#define WS_XT   1000000u  // [1000000,1400000)  x_in transposed to (N,4)

// ---- zero the atomic accumulators (S, count, U): floats [100000,1000000) ---
__global__ void bhs_zero(float* __restrict__ ws) {
  unsigned i = blockIdx.x * blockDim.x + threadIdx.x;
  if (i < 225000u) ((float4*)(ws + WS_S))[i] = make_float4(0.f, 0.f, 0.f, 0.f);
}

// ---- trace bookkeeping + effective_w + x transpose (one pass over N) -------
__global__ void bhs_traces(const float* __restrict__ x_in,
                           const float* __restrict__ post_spk,
                           const float* __restrict__ w_hat,
                           const float* __restrict__ s_e,
                           const float* __restrict__ e_trace,
                           const float* __restrict__ pre_trace,
                           const float* __restrict__ post_trace,
                           float* __restrict__ out,
                           float* __restrict__ ws) {
  unsigned i = blockIdx.x * blockDim.x + threadIdx.x;
  if (i >= NUM_NEURONS) return;
  const float decay = 0.95122942450071400910f;  // exp(-1/20)
  float x0 = x_in[i];
  float x1 = x_in[NUM_NEURONS + i];
  float x2 = x_in[2u * NUM_NEURONS + i];
  float x3 = x_in[3u * NUM_NEURONS + i];
  float pre_spk  = (x0 + x1 + x2 + x3) * 0.25f;
  float pre_new  = fmaf(pre_trace[i],  decay, pre_spk);
  float post_new = fmaf(post_trace[i], decay, post_spk[i]);
  float e_new    = fmaf(0.9f, e_trace[i], pre_spk * (post_new + 0.1f));
  const size_t OUTB = (size_t)BATCH * NUM_NEURONS * OUT_CH;  // 6,400,000
  out[OUTB + i]                 = e_new;
  out[OUTB + MAX_EDGES + i]     = pre_new;
  out[OUTB + 2u * MAX_EDGES + i] = post_new;
  ws[WS_EFFW + i] = s_e[i] * w_hat[i];
  ((float4*)(ws + WS_XT))[i] = make_float4(x0, x1, x2, x3);
}

// ---- pass A: node -> edge scatter (S, count) -------------------------------
__global__ void bhs_scatterA(const int* __restrict__ he, float* __restrict__ ws) {
  unsigned c = blockIdx.x * blockDim.x + threadIdx.x;  // exact grid, no guard
  int n = he[c];
  int e = he[NUM_CONN + c];
  float cw = ws[WS_EFFW + e];
  float4 xv = ((const float4*)(ws + WS_XT))[n];
  float* S = ws + WS_S + (size_t)e * 4u;
  atomicAdd(&S[0], xv.x * cw);
  atomicAdd(&S[1], xv.y * cw);
  atomicAdd(&S[2], xv.z * cw);
  atomicAdd(&S[3], xv.w * cw);
  atomicAdd(ws + WS_CNT + e, 1.0f);
}

// ---- per-edge mean: T = S / max(count,1), in place -------------------------
__global__ void bhs_norm(float* __restrict__ ws) {
  unsigned e = blockIdx.x * blockDim.x + threadIdx.x;
  if (e >= MAX_EDGES) return;
  float inv = 1.0f / fmaxf(ws[WS_CNT + e], 1.0f);
  float4* S = (float4*)(ws + WS_S);
  float4 s = S[e];
  s.x *= inv; s.y *= inv; s.z *= inv; s.w *= inv;
  S[e] = s;
}

// ---- pass B: edge -> node scatter (U) --------------------------------------
__global__ void bhs_scatterB(const int* __restrict__ he, float* __restrict__ ws) {
  unsigned c = blockIdx.x * blockDim.x + threadIdx.x;
  int n = he[c];
  int e = he[NUM_CONN + c];
  float cw = ws[WS_EFFW + e];
  float4 t = ((const float4*)(ws + WS_S))[e];
  float* U = ws + WS_U + (size_t)n * 4u;
  atomicAdd(&U[0], cw * t.x);
  atomicAdd(&U[1], cw * t.y);
  atomicAdd(&U[2], cw * t.z);
  atomicAdd(&U[3], cw * t.w);
}

// ---- epilogue: out[b,n,o] = U[n,b]*W[o] + bias[o] via WMMA -----------------
// One wave per 16x16 tile: M = 16 flattened (n*4+b) values, N = 16 channels.
// A: K=0 column holds U values (lanes 0-15, vgpr0); rest zero.
// B: K=0 row holds W[o]      (lanes 0-15, vgpr0); rest zero.
// C: bias[o] replicated across all 8 accumulator VGPRs.
__global__ void bhs_out_wmma(const float* __restrict__ weight_node,
                             const float* __restrict__ bias,
                             const float* __restrict__ ws,
                             float* __restrict__ out) {
  unsigned lane = threadIdx.x & 31u;
  unsigned wave = (blockIdx.x * blockDim.x + threadIdx.x) >> 5;  // 25,000 waves exact
  unsigned base = wave * 16u;                                    // first nb of tile
  unsigned oc = lane & 15u;
  float wv = weight_node[oc];
  float bv = bias[oc];
  const float* U = ws + WS_U;

  v2f a;
  a.x = (lane < 16u) ? U[base + lane] : 0.0f;   // A[M=lane, K=0]
  a.y = 0.0f;
  v2f b;
  b.x = (lane < 16u) ? wv : 0.0f;               // B[K=0, N=lane]
  b.y = 0.0f;
  v8f c;
#pragma unroll
  for (int v = 0; v < 8; ++v) c[v] = bv;

  v8f d = __builtin_amdgcn_wmma_f32_16x16x4_f32(
      /*neg_a=*/false, a, /*neg_b=*/false, b,
      /*c_mod=*/(short)0, c, /*reuse_a=*/false, /*reuse_b=*/false);

  unsigned mbase = (lane < 16u) ? 0u : 8u;      // D: vgpr v -> M=v (lo) / v+8 (hi)
#pragma unroll
  for (int v = 0; v < 8; ++v) {
    unsigned nb = base + mbase + (unsigned)v;   // nb = n*4 + b
    unsigned n = nb >> 2;
    unsigned bb = nb & 3u;
    out[((size_t)bb * NUM_NEURONS + n) * OUT_CH + oc] = d[v];
  }
}

extern "C" void kernel_launch(void* const* d_in, const int* in_sizes, int n_in,
                              void* d_out, int out_size, void* d_ws, size_t ws_size,
                              hipStream_t stream) {
  const float* x_in        = (const float*)d_in[0];
  const int*   he          = (const int*)d_in[1];   // harness: integer -> const int*
  const float* post_spk    = (const float*)d_in[2];
  const float* w_hat       = (const float*)d_in[3];
  const float* s_e         = (const float*)d_in[4];
  const float* e_trace     = (const float*)d_in[5];
  const float* pre_trace   = (const float*)d_in[6];
  const float* post_trace  = (const float*)d_in[7];
  const float* weight_node = (const float*)d_in[8];
  const float* bias        = (const float*)d_in[9];
  float* out = (float*)d_out;
  float* ws  = (float*)d_ws;

  bhs_zero<<<(225000u + 255u) / 256u, 256, 0, stream>>>(ws);
  bhs_traces<<<(NUM_NEURONS + 255u) / 256u, 256, 0, stream>>>(
      x_in, post_spk, w_hat, s_e, e_trace, pre_trace, post_trace, out, ws);
  bhs_scatterA<<<NUM_CONN / 256u, 256, 0, stream>>>(he, ws);
  bhs_norm<<<(MAX_EDGES + 255u) / 256u, 256, 0, stream>>>(ws);
  bhs_scatterB<<<NUM_CONN / 256u, 256, 0, stream>>>(he, ws);
  // 400,000 (n,b) rows / 16 = 25,000 tiles = 25,000 waves = 3125 blocks of 256
  bhs_out_wmma<<<3125, 256, 0, stream>>>(weight_node, bias, ws, out);
}